// GraphSAGE_80049600463399
// MI455X (gfx1250) — compile-verified
//
#include <hip/hip_runtime.h>

// GraphSAGE 2-layer (mean aggregator) for MI455X / gfx1250.
//
// Roofline reasoning:
//  * Aggregation: E=1.6M edges * 128 f32 -> ~205M global fp32 atomics/layer into a
//    51.2MB accumulator. Both x (51.2MB) and the accumulator fit in the 192MB L2,
//    so this is an L2-resident scatter: wave-per-edge, float4 gather, 4 no-return
//    global_atomic_add_f32 per lane.
//  * GEMM: 4 x (1e5x128 @ 128x128) = 13 GFLOP total -> negligible vs WMMA peak,
//    so keep full fp32 precision with V_WMMA_F32_16X16X4_F32 rather than bf16.
//    Fused epilogue: + bias, ReLU. Weights (64KB) stay cache-hot.

typedef __attribute__((ext_vector_type(2))) float v2f;
typedef __attribute__((ext_vector_type(8))) float v8f;

#define DIM 128
#define LDS_STRIDE 132   // pad 16x128 tile rows -> conflict-free b64 LDS reads

// ---------------------------------------------------------------- degrees ----
__global__ void sage_deg_kernel(const int* __restrict__ dst, float* __restrict__ deg, int E) {
    int e = blockIdx.x * blockDim.x + threadIdx.x;
    if (e < E) atomicAdd(&deg[dst[e]], 1.0f);
}

__global__ void sage_invdeg_kernel(float* __restrict__ deg, int n) {
    int i = blockIdx.x * blockDim.x + threadIdx.x;
    if (i < n) deg[i] = 1.0f / fmaxf(deg[i], 1.0f);
}

// --------------------------------------------------- edge scatter (sum) -----
// One wave (32 lanes) per edge; each lane moves 4 contiguous floats.
__global__ void sage_scatter_kernel(const float* __restrict__ x,
                                    const int* __restrict__ src,
                                    const int* __restrict__ dst,
                                    float* __restrict__ msg, int E) {
    long long t = (long long)blockIdx.x * blockDim.x + threadIdx.x;
    int e    = (int)(t >> 5);
    int lane = (int)(t & 31);
    if (e >= E) return;
    int s = src[e];
    int d = dst[e];
    const float4 v = ((const float4*)(x + (size_t)s * DIM))[lane];
    float* mp = msg + (size_t)d * DIM + lane * 4;
    atomicAdd(mp + 0, v.x);
    atomicAdd(mp + 1, v.y);
    atomicAdd(mp + 2, v.z);
    atomicAdd(mp + 3, v.w);
}

// ------------------------------------------------- fused SAGE layer GEMM ----
// out[i,:] = relu( xs[i,:] @ Wself + (msg[i,:]*invdeg[i]) @ Wneigh + bias )
// Block = 256 threads = 8 waves = one 16-row tile (N=100000 = 6250*16 exactly).
// Wave w computes output columns [16w, 16w+16) via chained f32 16x16x4 WMMAs.
// In-place safe for layer 2: each block reads only its own 16 rows of xs==out.
__launch_bounds__(256)
__global__ void sage_gemm_kernel(const float* __restrict__ xs,
                                 const float* __restrict__ msg,
                                 const float* __restrict__ invdeg,
                                 const float* __restrict__ Wself,
                                 const float* __restrict__ Wneigh,
                                 const float* __restrict__ bias,
                                 float* __restrict__ out) {
    __shared__ float As[16 * LDS_STRIDE];  // self rows
    __shared__ float An[16 * LDS_STRIDE];  // neighbor-mean rows

    const int row0 = blockIdx.x * 16;
    const int tid  = threadIdx.x;

    // Cooperative stage: 16x128 floats each, 8 floats (2 x float4) per thread.
    {
        int base = tid * 8;        // [0, 2048)
        int r    = base >> 7;      // row within tile
        int c    = base & 127;     // column
        const float4* sx = (const float4*)(xs  + (size_t)(row0 + r) * DIM + c);
        const float4* sm = (const float4*)(msg + (size_t)(row0 + r) * DIM + c);
        float4 a0 = sx[0], a1 = sx[1];
        float4 m0 = sm[0], m1 = sm[1];
        float sc = invdeg[row0 + r];
        m0.x *= sc; m0.y *= sc; m0.z *= sc; m0.w *= sc;
        m1.x *= sc; m1.y *= sc; m1.z *= sc; m1.w *= sc;
        float* ap = As + r * LDS_STRIDE + c;
        float* np = An + r * LDS_STRIDE + c;
        *(float4*)(ap)     = a0;  *(float4*)(ap + 4) = a1;
        *(float4*)(np)     = m0;  *(float4*)(np + 4) = m1;
    }
    __syncthreads();

    const int wave  = tid >> 5;      // 0..7 -> output column tile
    const int lane  = tid & 31;
    const int laneM = lane & 15;     // A: row M / B,D: column N
    const int kh    = lane >> 4;     // K-half select (lanes 16-31 hold K={2,3})
    const int col0  = wave * 16;

    v8f acc = {0.f, 0.f, 0.f, 0.f, 0.f, 0.f, 0.f, 0.f};

    const float* As_row = As + laneM * LDS_STRIDE;
    const float* An_row = An + laneM * LDS_STRIDE;

    // Self term: A = As tile, B = Wself[128x128 row-major]
#pragma unroll
    for (int k = 0; k < DIM / 4; ++k) {
        const int kk = 4 * k + 2 * kh;
        v2f a = *(const v2f*)(As_row + kk);            // A[M][kk], A[M][kk+1]
        v2f b;
        b.x = Wself[(size_t)kk * DIM + col0 + laneM];  // B[kk][N]
        b.y = Wself[(size_t)(kk + 1) * DIM + col0 + laneM];
        acc = __builtin_amdgcn_wmma_f32_16x16x4_f32(false, a, false, b,
                                                    (short)0, acc, false, false);
    }
    // Neighbor term: A = An tile, B = Wneigh
#pragma unroll
    for (int k = 0; k < DIM / 4; ++k) {
        const int kk = 4 * k + 2 * kh;
        v2f a = *(const v2f*)(An_row + kk);
        v2f b;
        b.x = Wneigh[(size_t)kk * DIM + col0 + laneM];
        b.y = Wneigh[(size_t)(kk + 1) * DIM + col0 + laneM];
        acc = __builtin_amdgcn_wmma_f32_16x16x4_f32(false, a, false, b,
                                                    (short)0, acc, false, false);
    }

    // Epilogue: bias + ReLU, scatter per C/D layout:
    // VGPR r, lanes 0-15 -> (M=r, N=lane); lanes 16-31 -> (M=8+r, N=lane-16).
    const float bb = bias[col0 + laneM];
    const int rbase = row0 + 8 * kh;
#pragma unroll
    for (int r = 0; r < 8; ++r) {
        float v = acc[r] + bb;
        v = fmaxf(v, 0.0f);
        out[(size_t)(rbase + r) * DIM + col0 + laneM] = v;
    }
}

// ----------------------------------------------------------------- driver ---
extern "C" void kernel_launch(void* const* d_in, const int* in_sizes, int n_in,
                              void* d_out, int out_size, void* d_ws, size_t ws_size,
                              hipStream_t stream) {
    const float* x   = (const float*)d_in[0];
    const int*   src = (const int*)  d_in[1];
    const int*   dst = (const int*)  d_in[2];
    const float* Ws1 = (const float*)d_in[3];
    const float* Wn1 = (const float*)d_in[4];
    const float* b1  = (const float*)d_in[5];
    const float* Ws2 = (const float*)d_in[6];
    const float* Wn2 = (const float*)d_in[7];
    const float* b2  = (const float*)d_in[8];
    float* out = (float*)d_out;

    const int N = in_sizes[0] / DIM;   // 100000
    const int E = in_sizes[1];         // 1600000

    float* msg    = (float*)d_ws;                   // N*DIM f32 accumulator
    float* invdeg = msg + (size_t)N * DIM;          // N f32

    const int threads = 256;
    const unsigned degBlocks  = (unsigned)((E + threads - 1) / threads);
    const unsigned nodeBlocks = (unsigned)((N + threads - 1) / threads);
    const unsigned scatBlocks = (unsigned)(((long long)E * 32 + threads - 1) / threads);
    const unsigned gemmBlocks = (unsigned)(N / 16); // N is a multiple of 16

    // Degrees (graph-only; shared by both layers)
    hipMemsetAsync(invdeg, 0, (size_t)N * sizeof(float), stream);
    sage_deg_kernel<<<degBlocks, threads, 0, stream>>>(dst, invdeg, E);
    sage_invdeg_kernel<<<nodeBlocks, threads, 0, stream>>>(invdeg, N);

    // ---- Layer 1: aggregate x, fused GEMM -> out (h1) ----
    hipMemsetAsync(msg, 0, (size_t)N * DIM * sizeof(float), stream);
    sage_scatter_kernel<<<scatBlocks, threads, 0, stream>>>(x, src, dst, msg, E);
    sage_gemm_kernel<<<gemmBlocks, threads, 0, stream>>>(x, msg, invdeg, Ws1, Wn1, b1, out);

    // ---- Layer 2: aggregate h1 (in d_out), fused GEMM in-place -> out ----
    hipMemsetAsync(msg, 0, (size_t)N * DIM * sizeof(float), stream);
    sage_scatter_kernel<<<scatBlocks, threads, 0, stream>>>(out, src, dst, msg, E);
    sage_gemm_kernel<<<gemmBlocks, threads, 0, stream>>>(out, msg, invdeg, Ws2, Wn2, b2, out);
}